// DilatedGCN_38448547233862
// MI455X (gfx1250) — compile-verified
//
#include <hip/hip_runtime.h>
#include <math.h>

// DilatedGCN forward for MI455X (gfx1250), wave32, WMMA f32 16x16x4.
// All math in f32 (top-k index selection is precision-sensitive). The
// normalized adjacency has uniform degree 21 -> aggregation is a 21-way
// gather, no dense bmm. GEMMs: each wave computes a 16x64 tile (4 WMMA
// accumulators reuse one A fragment); W panels staged through LDS.

#define BATCH 32
#define NPTS  1024
#define KNN   20
#define BN_EPS 1e-5f
#define MROWS_TOTAL (BATCH * NPTS)

typedef __attribute__((ext_vector_type(2))) float v2f;
typedef __attribute__((ext_vector_type(8))) float v8f;

// ---------------- squared norms ----------------
template <int C>
__global__ void sq_norm_kernel(const float* __restrict__ feats, float* __restrict__ sq) {
  int i = blockIdx.x * blockDim.x + threadIdx.x;
  if (i >= MROWS_TOTAL) return;
  const float* p = feats + (size_t)i * C;
  float s = 0.f;
#pragma unroll
  for (int c = 0; c < C; ++c) s += p[c] * p[c];
  sq[i] = s;
}

// ---------------- pad (B*N,3) -> (B*N,4) with zero ----------------
__global__ void pad3to4_kernel(const float* __restrict__ x, float* __restrict__ xp) {
  int i = blockIdx.x * blockDim.x + threadIdx.x;
  if (i >= MROWS_TOTAL) return;
  xp[(size_t)i * 4 + 0] = x[(size_t)i * 3 + 0];
  xp[(size_t)i * 4 + 1] = x[(size_t)i * 3 + 1];
  xp[(size_t)i * 4 + 2] = x[(size_t)i * 3 + 2];
  xp[(size_t)i * 4 + 3] = 0.f;
}

// ---------------- fused dist^2 + dilated kNN selection ----------------
// One wave per point-row; each lane owns columns m = j*32 + lane. Repeated
// wave-wide argmin with lexicographic (value, index) compare reproduces
// top_k ordering incl. tie-break by lowest index; winner evicted to +inf.
template <int C>
__global__ void knn_kernel(const float* __restrict__ feats, const float* __restrict__ sq,
                           int* __restrict__ nbr, int dilation) {
  const int wave = threadIdx.x >> 5;
  const int lane = threadIdx.x & 31;
  const int rowg = blockIdx.x * 8 + wave;
  const int b = rowg >> 10;
  const int n = rowg & (NPTS - 1);
  __shared__ float rowbuf[8][C];
  const float* fb = feats + (size_t)b * NPTS * C;
  for (int c = lane; c < C; c += 32) rowbuf[wave][c] = fb[(size_t)n * C + c];
  __syncthreads();

  const float sqn = sq[rowg];
  const float* sqb = sq + b * NPTS;
  float dloc[32];
#pragma unroll 4
  for (int j = 0; j < 32; ++j) {
    const int m = j * 32 + lane;
    const float* pm = fb + (size_t)m * C;
    float dot = 0.f;
#pragma unroll
    for (int c = 0; c < C; ++c) dot += rowbuf[wave][c] * pm[c];
    dloc[j] = sqn + sqb[m] - 2.f * dot;   // exact reference formula
  }

  const float FINF = 3.0e38f;
  int sel = 0;
  const int maxT = 2 + (KNN - 1) * dilation;  // need positions 0..1+(K-1)*d
  for (int t = 0; t < maxT; ++t) {
    float bv = FINF;
    int bi = 0x7fffffff;
#pragma unroll
    for (int j = 0; j < 32; ++j) {
      const int m = j * 32 + lane;
      if (dloc[j] < bv) { bv = dloc[j]; bi = m; }
    }
#pragma unroll
    for (int off = 16; off > 0; off >>= 1) {
      float ov = __shfl_xor(bv, off, 32);
      int   oi = __shfl_xor(bi, off, 32);
      if (ov < bv || (ov == bv && oi < bi)) { bv = ov; bi = oi; }
    }
#pragma unroll
    for (int j = 0; j < 32; ++j)
      if (j * 32 + lane == bi) dloc[j] = FINF;
    if (t >= 1 && ((t - 1) % dilation) == 0 && sel < KNN) {
      if (lane == 0) nbr[(size_t)rowg * KNN + sel] = bi;
      ++sel;
    }
  }
}

// ---------------- WMMA f32 GEMM, K compile-time, 16x64 tile per wave ----------------
// out = A(MxK) @ W(KxN) + bias, epilogue per EPI:
//   EPI 0: bias only          EPI 1: bias+BN+ReLU store
//   EPI 2: bias+BN+ReLU then atomic-max pool into out[(row/NPTS)*N+col]
// Block = 256 threads = 8 waves. Block covers 64 columns (4 col-tiles) x
// 128 rows (8 row-tiles). W panel staged in LDS in K-chunks of <=256
// (<=64 KB), pair-interleaved per col-tile so B fragments are contiguous
// ds_load_b64 and both loads and stores are bank-conflict-free. One A
// fragment (global_load_b64) feeds 4 WMMAs.
template <int K, int EPI>
__global__ __launch_bounds__(256) void gemm_wmma(
    const float* __restrict__ A, const float* __restrict__ W,
    const float* __restrict__ bias,
    const float* __restrict__ bng, const float* __restrict__ bnb,
    const float* __restrict__ bnm, const float* __restrict__ bnv,
    float* __restrict__ out, int M, int N, int globalRowBase) {
  constexpr int KT = (K < 256) ? K : 256;     // LDS K-chunk
  __shared__ float ldsW[64 * KT];             // 4 sub-tiles of 16*KT
  const int tid  = threadIdx.x;
  const int lane = tid & 31;
  const int wave = tid >> 5;
  const int col0 = blockIdx.x << 6;           // 64 cols per block
  const int rowTile = blockIdx.y * 8 + wave;  // wave-uniform
  const bool active = (rowTile << 4) < M;
  const int half = lane >> 4;
  const int l16  = lane & 15;
  const int row0 = rowTile << 4;
  const float* Ap = A + (size_t)(row0 + l16) * K + 2 * half;

  v8f acc0 = {}, acc1 = {}, acc2 = {}, acc3 = {};
  for (int kb = 0; kb < K; kb += KT) {
    // cooperative W panel load; (k0, col-tile t, c) -> t*16*KT + (k0/2)*32 + 2*c + (k0&1)
    for (int idx = tid; idx < 64 * KT; idx += 256) {
      const int k0  = idx >> 6;
      const int c64 = idx & 63;               // coalesced along columns
      const int t   = c64 >> 4;
      const int c   = c64 & 15;
      const int colg = col0 + c64;
      const int colc = colg < N ? colg : N - 1;   // clamp; discarded at store
      ldsW[t * (16 * KT) + (k0 >> 1) * 32 + 2 * c + (k0 & 1)] =
          W[(size_t)(kb + k0) * N + colc];
    }
    __syncthreads();
    if (active) {
#pragma unroll 4
      for (int kk = 0; kk < KT; kk += 4) {
        const int bofs = ((kk >> 1) + half) * 32 + 2 * l16;
        v2f av = *reinterpret_cast<const v2f*>(Ap + kb + kk);
        v2f b0 = *reinterpret_cast<const v2f*>(&ldsW[0 * (16 * KT) + bofs]);
        v2f b1 = *reinterpret_cast<const v2f*>(&ldsW[1 * (16 * KT) + bofs]);
        v2f b2 = *reinterpret_cast<const v2f*>(&ldsW[2 * (16 * KT) + bofs]);
        v2f b3 = *reinterpret_cast<const v2f*>(&ldsW[3 * (16 * KT) + bofs]);
        acc0 = __builtin_amdgcn_wmma_f32_16x16x4_f32(false, av, false, b0, (short)0, acc0, false, false);
        acc1 = __builtin_amdgcn_wmma_f32_16x16x4_f32(false, av, false, b1, (short)0, acc1, false, false);
        acc2 = __builtin_amdgcn_wmma_f32_16x16x4_f32(false, av, false, b2, (short)0, acc2, false, false);
        acc3 = __builtin_amdgcn_wmma_f32_16x16x4_f32(false, av, false, b3, (short)0, acc3, false, false);
      }
    }
    __syncthreads();
  }

  if (active) {
    v8f accs[4] = {acc0, acc1, acc2, acc3};
#pragma unroll
    for (int t = 0; t < 4; ++t) {
      const int col = col0 + t * 16 + l16;
      if (col < N) {
        const float bia = bias[col];
        float g = 1.f, be = 0.f, mu = 0.f, iv = 1.f;
        if (EPI >= 1) { g = bng[col]; be = bnb[col]; mu = bnm[col]; iv = rsqrtf(bnv[col] + BN_EPS); }
#pragma unroll
        for (int r = 0; r < 8; ++r) {
          const int row = row0 + r + 8 * half;  // C/D layout: lanes 16-31 carry rows +8
          float v = accs[t][r] + bia;
          if (EPI >= 1) { v = (v - mu) * iv * g + be; v = fmaxf(v, 0.f); }
          if (EPI == 2) {
            const int rowg = globalRowBase + row;
            const int bidx = rowg >> 10;        // / NPTS ; post-ReLU v>=0 -> int max exact
            atomicMax((int*)(out + (size_t)bidx * N + col), __float_as_int(v));
          } else {
            out[(size_t)row * N + col] = v;
          }
        }
      }
    }
  }
}

// ---------------- GCN aggregation: (self + 20 nbrs)/21, BN, ReLU ----------------
__global__ void gcn_agg_kernel(const float* __restrict__ h, const int* __restrict__ nbr,
                               const float* __restrict__ g, const float* __restrict__ be,
                               const float* __restrict__ mu, const float* __restrict__ var,
                               float* __restrict__ featsOut, float* __restrict__ msOut) {
  const int rowg = blockIdx.x;
  const int c = threadIdx.x;         // 64 channels
  const int b = rowg >> 10;
  __shared__ int nb[KNN];
  if (c < KNN) nb[c] = nbr[(size_t)rowg * KNN + c];
  __syncthreads();
  float s = h[(size_t)rowg * 64 + c];
  const float* hb = h + (size_t)b * NPTS * 64;
#pragma unroll 4
  for (int j = 0; j < KNN; ++j) s += hb[(size_t)nb[j] * 64 + c];
  s *= (1.f / 21.f);
  s = (s - mu[c]) * rsqrtf(var[c] + BN_EPS) * g[c] + be[c];
  s = fmaxf(s, 0.f);
  featsOut[(size_t)rowg * 64 + c] = s;
  msOut[(size_t)rowg * 64 + c] = s;
}

__global__ void mean3_kernel(const float* __restrict__ a, const float* __restrict__ b,
                             const float* __restrict__ c, float* __restrict__ o, size_t n) {
  size_t i = (size_t)blockIdx.x * blockDim.x + threadIdx.x;
  if (i < n) o[i] = (a[i] + b[i] + c[i]) * (1.f / 3.f);
}

// host-side templated launch helper (block covers 64 cols x 128 rows)
template <int K, int EPI>
static void launch_gemm(const float* A, const float* W, const float* bias,
                        const float* g, const float* b_, const float* m_, const float* v_,
                        float* out, int M, int N, int rowBase, hipStream_t stream) {
  dim3 grid((unsigned)((N + 63) / 64), (unsigned)((M / 16 + 7) / 8));
  gemm_wmma<K, EPI><<<grid, 256, 0, stream>>>(A, W, bias, g, b_, m_, v_, out, M, N, rowBase);
}

extern "C" void kernel_launch(void* const* d_in, const int* in_sizes, int n_in,
                              void* d_out, int out_size, void* d_ws, size_t ws_size,
                              hipStream_t stream) {
  (void)in_sizes; (void)n_in; (void)out_size; (void)ws_size;
  const float* x = (const float*)d_in[0];
  const float* Wg[3] = {(const float*)d_in[1], (const float*)d_in[7],  (const float*)d_in[13]};
  const float* bg[3] = {(const float*)d_in[2], (const float*)d_in[8],  (const float*)d_in[14]};
  const float* gg[3] = {(const float*)d_in[3], (const float*)d_in[9],  (const float*)d_in[15]};
  const float* gb[3] = {(const float*)d_in[4], (const float*)d_in[10], (const float*)d_in[16]};
  const float* gm[3] = {(const float*)d_in[5], (const float*)d_in[11], (const float*)d_in[17]};
  const float* gv[3] = {(const float*)d_in[6], (const float*)d_in[12], (const float*)d_in[18]};
  const float *W1=(const float*)d_in[19], *b1=(const float*)d_in[20];
  const float *n1g=(const float*)d_in[21], *n1b=(const float*)d_in[22], *n1m=(const float*)d_in[23], *n1v=(const float*)d_in[24];
  const float *W2=(const float*)d_in[25], *b2=(const float*)d_in[26];
  const float *n2g=(const float*)d_in[27], *n2b=(const float*)d_in[28], *n2m=(const float*)d_in[29], *n2v=(const float*)d_in[30];
  const float *W3=(const float*)d_in[31], *b3=(const float*)d_in[32];
  const float *n3g=(const float*)d_in[33], *n3b=(const float*)d_in[34], *n3m=(const float*)d_in[35], *n3v=(const float*)d_in[36];
  const float *W4=(const float*)d_in[37], *b4=(const float*)d_in[38];
  const float *n4g=(const float*)d_in[39], *n4b=(const float*)d_in[40], *n4m=(const float*)d_in[41], *n4v=(const float*)d_in[42];
  const float *W5=(const float*)d_in[43], *b5=(const float*)d_in[44];
  const float *n5g=(const float*)d_in[45], *n5b=(const float*)d_in[46], *n5m=(const float*)d_in[47], *n5v=(const float*)d_in[48];
  const float *W6=(const float*)d_in[49], *b6=(const float*)d_in[50];

  // workspace layout (~65 MB of f32)
  float* wsf = (float*)d_ws;
  size_t o = 0;
  const size_t FEAT = (size_t)MROWS_TOTAL * 64;
  float* feats = wsf + o; o += FEAT;
  float* hbuf  = wsf + o; o += FEAT;
  float* m0    = wsf + o; o += FEAT;
  float* m1    = wsf + o; o += FEAT;
  float* m2    = wsf + o; o += FEAT;
  float* fmean = wsf + o; o += FEAT;
  float* sq    = wsf + o; o += (size_t)MROWS_TOTAL;
  float* xp    = wsf + o; o += (size_t)MROWS_TOTAL * 4;   // padded layer-0 input
  float* f1c   = wsf + o; o += (size_t)4096 * 128;
  float* f2c   = wsf + o; o += (size_t)4096 * 512;
  float* pooled= wsf + o; o += (size_t)BATCH * 1024;
  float* f4    = wsf + o; o += (size_t)BATCH * 1024;
  float* f5    = wsf + o; o += (size_t)BATCH * 512;
  int*   nbr   = (int*)(wsf + o);

  // ---- 3 dilated GCN layers ----
  const int dil[3] = {1, 2, 4};
  float* mouts[3] = {m0, m1, m2};
  pad3to4_kernel<<<(MROWS_TOTAL + 255) / 256, 256, 0, stream>>>(x, xp);
  for (int l = 0; l < 3; ++l) {
    if (l == 0) {
      sq_norm_kernel<3><<<(MROWS_TOTAL + 255) / 256, 256, 0, stream>>>(x, sq);
      knn_kernel<3><<<MROWS_TOTAL / 8, 256, 0, stream>>>(x, sq, nbr, dil[l]);
      launch_gemm<4, 0>(xp, Wg[l], bg[l], nullptr, nullptr, nullptr, nullptr,
                        hbuf, MROWS_TOTAL, 64, 0, stream);
    } else {
      sq_norm_kernel<64><<<(MROWS_TOTAL + 255) / 256, 256, 0, stream>>>(feats, sq);
      knn_kernel<64><<<MROWS_TOTAL / 8, 256, 0, stream>>>(feats, sq, nbr, dil[l]);
      launch_gemm<64, 0>(feats, Wg[l], bg[l], nullptr, nullptr, nullptr, nullptr,
                         hbuf, MROWS_TOTAL, 64, 0, stream);
    }
    gcn_agg_kernel<<<MROWS_TOTAL, 64, 0, stream>>>(hbuf, nbr, gg[l], gb[l], gm[l], gv[l],
                                                   feats, mouts[l]);
  }

  // ---- multi-scale mean ----
  mean3_kernel<<<(int)((FEAT + 255) / 256), 256, 0, stream>>>(m0, m1, m2, fmean, FEAT);

  // ---- MLP (row-chunked); max-pool fused into the W3 GEMM via atomicMax ----
  hipMemsetAsync(pooled, 0, (size_t)BATCH * 1024 * sizeof(float), stream);
  for (int ch = 0; ch < 8; ++ch) {
    int base = ch * 4096;
    launch_gemm<64, 1>(fmean + (size_t)base * 64, W1, b1, n1g, n1b, n1m, n1v,
                       f1c, 4096, 128, 0, stream);
    launch_gemm<128, 1>(f1c, W2, b2, n2g, n2b, n2m, n2v, f2c, 4096, 512, 0, stream);
    launch_gemm<512, 2>(f2c, W3, b3, n3g, n3b, n3m, n3v, pooled, 4096, 1024, base, stream);
  }
  launch_gemm<1024, 1>(pooled, W4, b4, n4g, n4b, n4m, n4v, f4, BATCH, 1024, 0, stream);
  launch_gemm<1024, 1>(f4, W5, b5, n5g, n5b, n5m, n5v, f5, BATCH, 512, 0, stream);
  launch_gemm<512, 0>(f5, W6, b6, nullptr, nullptr, nullptr, nullptr,
                      (float*)d_out, BATCH, 40, 0, stream);
}